// GlyphClassifier_20263655703049
// MI455X (gfx1250) — compile-verified
//
#include <hip/hip_runtime.h>

// ---------------------------------------------------------------------------
// GlyphClassifier for MI455X (gfx1250, wave32).
// Phase 1: ragged mean-pool over hidden_states[b, start:end] + LayerNorm(g2,b2)
//          -> sel[B,H], nx[B]          (reads only the masked rows, ~25 MB)
// Phase 2: t = glyph_pooled @ W^T + b  via V_WMMA_F32_16X16X4_F32 (exact fp32)
// Phase 3: per-row LayerNorm(g1,b1) of t fused with cosine vs sel -> score[B,S]
// ---------------------------------------------------------------------------

#define BB 64
#define LL 512
#define HH 768
#define SS 32

typedef float v2f __attribute__((ext_vector_type(2)));
typedef float v8f __attribute__((ext_vector_type(8)));

__device__ __forceinline__ float blockReduceSum(float v, float* sm) {
    int t = threadIdx.x;
    sm[t] = v;
    __syncthreads();
    for (int s = 128; s > 0; s >>= 1) {
        if (t < s) sm[t] += sm[t + s];
        __syncthreads();
    }
    float r = sm[0];
    __syncthreads();
    return r;
}

// ---------------- Phase 1: ragged mean pool + LayerNorm(g2,b2) -------------
__global__ __launch_bounds__(256) void glyph_pool_ln(
    const float* __restrict__ hs, const int* __restrict__ startA,
    const int* __restrict__ endA, const float* __restrict__ g2,
    const float* __restrict__ b2, float* __restrict__ sel,
    float* __restrict__ nxA) {
    __shared__ float sm[256];
    const int b = blockIdx.x;
    const int t = threadIdx.x;
    const int s = startA[b];
    const int e = endA[b];

    float a0 = 0.f, a1 = 0.f, a2 = 0.f;
    const float* base = hs + (size_t)b * LL * HH;
    for (int l = s; l < e; ++l) {
        const float* row = base + (size_t)l * HH;
        a0 += row[t];
        a1 += row[t + 256];
        a2 += row[t + 512];
    }
    int n = e - s;
    float inv = 1.0f / (float)(n < 1 ? 1 : n);
    a0 *= inv; a1 *= inv; a2 *= inv;            // seg_mean elements

    float mu  = blockReduceSum(a0 + a1 + a2, sm) * (1.0f / 768.0f);
    float d0 = a0 - mu, d1 = a1 - mu, d2 = a2 - mu;
    float var = blockReduceSum(d0 * d0 + d1 * d1 + d2 * d2, sm) * (1.0f / 768.0f);
    float rs  = rsqrtf(var + 1e-12f);

    float y0 = d0 * rs * g2[t]       + b2[t];
    float y1 = d1 * rs * g2[t + 256] + b2[t + 256];
    float y2 = d2 * rs * g2[t + 512] + b2[t + 512];
    sel[b * HH + t]       = y0;
    sel[b * HH + t + 256] = y1;
    sel[b * HH + t + 512] = y2;

    float nsq = blockReduceSum(y0 * y0 + y1 * y1 + y2 * y2, sm);
    if (t == 0) nxA[b] = sqrtf(nsq);
}

// ---------------- Phase 2: fp32 WMMA GEMM  t = A @ W^T + bias --------------
// One wave32 per 32x32 output tile: 2x2 WMMA sub-tiles, K stepped by 4.
// A-frag (16x4 f32): lane holds A[m0+lane%16][k + 2*(lane>>4) + {0,1}]
// B-frag (4x16 f32): B[k][n] = W[n][k]; lane holds W[n0+lane%16][k+2*(lane>>4)+{0,1}]
// C/D (16x16 f32):   vgpr r, lane -> (M = r + 8*(lane>>4), N = lane%16)
__global__ __launch_bounds__(256) void glyph_gemm_wmma(
    const float* __restrict__ A, const float* __restrict__ W,
    const float* __restrict__ bias, float* __restrict__ T) {
    const int wave = (blockIdx.x * blockDim.x + threadIdx.x) >> 5;
    const int lane = threadIdx.x & 31;
    const int mTile = wave & 63;   // 2048/32 = 64 tiles in M
    const int nTile = wave >> 6;   //  768/32 = 24 tiles in N
    const int m0 = mTile * 32;
    const int n0 = nTile * 32;
    const int laneM  = lane & 15;
    const int laneHi = lane >> 4;  // selects K pair {0,1} vs {2,3}

    const float* a0p = A + (size_t)(m0 + laneM)      * HH + 2 * laneHi;
    const float* a1p = A + (size_t)(m0 + 16 + laneM) * HH + 2 * laneHi;
    const float* b0p = W + (size_t)(n0 + laneM)      * HH + 2 * laneHi;
    const float* b1p = W + (size_t)(n0 + 16 + laneM) * HH + 2 * laneHi;

    const float bc0 = bias[n0 + laneM];
    const float bc1 = bias[n0 + 16 + laneM];
    v8f acc00, acc01, acc10, acc11;
#pragma unroll
    for (int r = 0; r < 8; ++r) {
        acc00[r] = bc0; acc01[r] = bc1; acc10[r] = bc0; acc11[r] = bc1;
    }

    for (int k = 0; k < HH; k += 4) {
        v2f a0 = *(const v2f*)(a0p + k);
        v2f a1 = *(const v2f*)(a1p + k);
        v2f b0 = *(const v2f*)(b0p + k);
        v2f b1 = *(const v2f*)(b1p + k);
        if (k + 4 < HH) {
            // locality 3 -> WGP-scope global_prefetch_b8 (warms L0, not just GL2)
            __builtin_prefetch(a0p + k + 4, 0, 3);
            __builtin_prefetch(b0p + k + 4, 0, 3);
        }
        acc00 = __builtin_amdgcn_wmma_f32_16x16x4_f32(false, a0, false, b0,
                                                      (short)0, acc00, false, false);
        acc01 = __builtin_amdgcn_wmma_f32_16x16x4_f32(false, a0, false, b1,
                                                      (short)0, acc01, false, false);
        acc10 = __builtin_amdgcn_wmma_f32_16x16x4_f32(false, a1, false, b0,
                                                      (short)0, acc10, false, false);
        acc11 = __builtin_amdgcn_wmma_f32_16x16x4_f32(false, a1, false, b1,
                                                      (short)0, acc11, false, false);
    }

    const int colLo = n0 + laneM;
#pragma unroll
    for (int r = 0; r < 8; ++r) {
        const int rowLo = m0 + r + 8 * laneHi;
        T[(size_t)rowLo * HH + colLo]             = acc00[r];
        T[(size_t)rowLo * HH + colLo + 16]        = acc01[r];
        T[(size_t)(rowLo + 16) * HH + colLo]      = acc10[r];
        T[(size_t)(rowLo + 16) * HH + colLo + 16] = acc11[r];
    }
}

// ---------------- Phase 3: row LayerNorm(g1,b1) fused with cosine ----------
__global__ __launch_bounds__(256) void glyph_ln_cos(
    const float* __restrict__ T, const float* __restrict__ g1,
    const float* __restrict__ b1, const float* __restrict__ sel,
    const float* __restrict__ nxA, float* __restrict__ out) {
    __shared__ float sm[256];
    const int row = blockIdx.x;      // 0..B*S-1
    const int b   = row >> 5;        // /S
    const int t   = threadIdx.x;

    const float* x = T + (size_t)row * HH;
    float x0 = x[t], x1 = x[t + 256], x2 = x[t + 512];

    float mu  = blockReduceSum(x0 + x1 + x2, sm) * (1.0f / 768.0f);
    float d0 = x0 - mu, d1 = x1 - mu, d2 = x2 - mu;
    float var = blockReduceSum(d0 * d0 + d1 * d1 + d2 * d2, sm) * (1.0f / 768.0f);
    float rs  = rsqrtf(var + 1e-12f);

    float y0 = d0 * rs * g1[t]       + b1[t];
    float y1 = d1 * rs * g1[t + 256] + b1[t + 256];
    float y2 = d2 * rs * g1[t + 512] + b1[t + 512];

    const float* sb = sel + b * HH;
    float dot = blockReduceSum(sb[t] * y0 + sb[t + 256] * y1 + sb[t + 512] * y2, sm);
    float nsq = blockReduceSum(y0 * y0 + y1 * y1 + y2 * y2, sm);
    if (t == 0) {
        float denom = fmaxf(nxA[b] * sqrtf(nsq), 1e-8f);
        out[row] = dot / denom;
    }
}

// ---------------------------------------------------------------------------
extern "C" void kernel_launch(void* const* d_in, const int* in_sizes, int n_in,
                              void* d_out, int out_size, void* d_ws, size_t ws_size,
                              hipStream_t stream) {
    (void)in_sizes; (void)n_in; (void)out_size; (void)ws_size;
    const float* hidden = (const float*)d_in[0];   // [B, L, H]
    const float* gp     = (const float*)d_in[1];   // [B*S, H]
    const float* W      = (const float*)d_in[2];   // [H, H]
    const float* bias   = (const float*)d_in[3];   // [H]
    const float* g1     = (const float*)d_in[4];
    const float* b1     = (const float*)d_in[5];
    const float* g2     = (const float*)d_in[6];
    const float* b2     = (const float*)d_in[7];
    const int*   gstart = (const int*)d_in[8];     // [B]
    const int*   gend   = (const int*)d_in[9];     // [B]
    float* score = (float*)d_out;                  // [B, S]

    float* ws   = (float*)d_ws;
    float* sel  = ws;                       // B*H floats
    float* nx   = ws + BB * HH;             // B floats
    float* tbuf = ws + BB * HH + BB;        // (B*S)*H floats

    glyph_pool_ln<<<BB, 256, 0, stream>>>(hidden, gstart, gend, g2, b2, sel, nx);

    // 64 * 24 = 1536 waves, 8 waves/block -> 192 blocks
    glyph_gemm_wmma<<<192, 256, 0, stream>>>(gp, W, bias, tbuf);

    glyph_ln_cos<<<BB * SS, 256, 0, stream>>>(tbuf, g1, b1, sel, nx, score);
}